// GAT_22797686407653
// MI455X (gfx1250) — compile-verified
//
#include <hip/hip_runtime.h>
#include <math.h>

// ---- problem constants (match reference) ----
#define NNODES 20000
#define NEDGES 320000
#define NGRAPH 64
#define DIMIN  128
#define DH     64
#define NH     8
#define NC     32
#define FDIM   (NH*DH)   // 512
#define NEGS   0.2f

typedef float v2f __attribute__((ext_vector_type(2)));
typedef float v8f __attribute__((ext_vector_type(8)));

// -------------------------------------------------------------------------
// fill
// -------------------------------------------------------------------------
__global__ void fill_f32(float* __restrict__ p, float v, int n) {
    int i = blockIdx.x * blockDim.x + threadIdx.x;
    if (i < n) p[i] = v;
}

// -------------------------------------------------------------------------
// out[M,N] = A[M,K] @ W[K,N] + bias[N]   (fp32 WMMA 16x16x4, wave-per-tile)
// A-frag layout (ISA 7.12.2): lanes 0-15 hold M=0..15, v0=K0,v1=K1;
//                             lanes 16-31 same M, v0=K2,v1=K3.
// B-frag mirrors with N in lanes; C/D: N=lane&15, VGPR r -> M=r (+8 if lane>=16)
// -------------------------------------------------------------------------
__global__ void gemm_wmma_f32(const float* __restrict__ A,
                              const float* __restrict__ W,
                              const float* __restrict__ bias,
                              float* __restrict__ out,
                              int M, int K, int Nn) {
    int lane = threadIdx.x & 31;
    int wave = threadIdx.x >> 5;
    int tilesN = Nn >> 4;
    int tiles  = (M >> 4) * tilesN;
    int tile = blockIdx.x * (blockDim.x >> 5) + wave;
    if (tile >= tiles) return;                 // wave-uniform: EXEC stays all-1s
    int tm = tile / tilesN, tn = tile % tilesN;

    int mrow  = tm * 16 + (lane & 15);
    int col   = tn * 16 + (lane & 15);
    int khalf = (lane >> 4) << 1;              // 0 or 2

    v8f c = {0.f,0.f,0.f,0.f,0.f,0.f,0.f,0.f};
    for (int k0 = 0; k0 < K; k0 += 4) {
        int kk = k0 + khalf;
        v2f a, b;
        a.x = A[(size_t)mrow * K + kk];
        a.y = A[(size_t)mrow * K + kk + 1];
        b.x = W[(size_t)kk * Nn + col];
        b.y = W[(size_t)(kk + 1) * Nn + col];
        c = __builtin_amdgcn_wmma_f32_16x16x4_f32(false, a, false, b,
                                                  (short)0, c, false, false);
    }
    float bcol  = bias[col];
    int   rbase = tm * 16 + ((lane >> 4) << 3);
#pragma unroll
    for (int r = 0; r < 8; r++)
        out[(size_t)(rbase + r) * Nn + col] = c[r] + bcol;
}

// -------------------------------------------------------------------------
// order-preserving float->uint encoding for atomic segment-max
// -------------------------------------------------------------------------
__device__ __forceinline__ unsigned f32_ord(float f) {
    unsigned u = __float_as_uint(f);
    return (u & 0x80000000u) ? ~u : (u | 0x80000000u);
}

// wave32 per edge: lanes 0-3 -> head0 ... lanes 28-31 -> head7, 16 dims/lane
__global__ void edge_scores(const float* __restrict__ fs,
                            const float* __restrict__ fd,
                            const float* __restrict__ attn,
                            const int* __restrict__ src,
                            const int* __restrict__ dst,
                            float* __restrict__ scores,
                            unsigned* __restrict__ mEnc) {
    int lane = threadIdx.x & 31;
    int e = blockIdx.x * 8 + (threadIdx.x >> 5);
    if (e >= NEDGES) return;
    int s = src[e], d = dst[e];
    int head = lane >> 2, sub = lane & 3;

    const float4* fs4 = (const float4*)fs;
    const float4* fd4 = (const float4*)fd;
    float partial = 0.f;
#pragma unroll
    for (int j = 0; j < 4; j++) {
        int f4i = head * 16 + sub * 4 + j;          // float4 index within 512
        float4 a = fs4[(size_t)s * 128 + f4i];
        float4 b = fd4[(size_t)d * 128 + f4i];
        const float* at = attn + head * DH + sub * 16 + j * 4;
        float x;
        x = a.x + b.x; partial += at[0] * ((x > 0.f) ? x : NEGS * x);
        x = a.y + b.y; partial += at[1] * ((x > 0.f) ? x : NEGS * x);
        x = a.z + b.z; partial += at[2] * ((x > 0.f) ? x : NEGS * x);
        x = a.w + b.w; partial += at[3] * ((x > 0.f) ? x : NEGS * x);
    }
    partial += __shfl_xor(partial, 1, 32);
    partial += __shfl_xor(partial, 2, 32);
    if (sub == 0) {
        scores[(size_t)e * NH + head] = partial;
        atomicMax(&mEnc[(size_t)d * NH + head], f32_ord(partial));
    }
}

// decode encoded max in place; enc==0 sentinel -> node with no in-edges -> 0
__global__ void decode_m(unsigned* __restrict__ m, int n) {
    int i = blockIdx.x * blockDim.x + threadIdx.x;
    if (i >= n) return;
    unsigned e = m[i];
    float f = 0.f;
    if (e != 0u)
        f = (e & 0x80000000u) ? __uint_as_float(e & 0x7FFFFFFFu)
                              : __uint_as_float(~e);
    ((float*)m)[i] = f;
}

// w = exp(score - m[dst]); denom[dst] += w   (thread per edge-head)
__global__ void edge_expw(float* __restrict__ scores,
                          const float* __restrict__ m,
                          float* __restrict__ denom,
                          const int* __restrict__ dst) {
    int t = blockIdx.x * blockDim.x + threadIdx.x;
    if (t >= NEDGES * NH) return;
    int e = t >> 3, h = t & 7;
    int d = dst[e];
    float w = expf(scores[t] - m[(size_t)d * NH + h]);
    scores[t] = w;
    atomicAdd(&denom[(size_t)d * NH + h], w);
}

// agg[dst,h,:] += fs[src,h,:] * alpha      (wave32 per edge)
__global__ void edge_agg(const float* __restrict__ fs,
                         const float* __restrict__ w,
                         const float* __restrict__ denom,
                         const int* __restrict__ src,
                         const int* __restrict__ dst,
                         float* __restrict__ agg) {
    int lane = threadIdx.x & 31;
    int e = blockIdx.x * 8 + (threadIdx.x >> 5);
    if (e >= NEDGES) return;
    int s = src[e], d = dst[e];
    float alpha = 0.f;
    if (lane < NH) {
        float ww = w[(size_t)e * NH + lane];
        float dn = denom[(size_t)d * NH + lane];
        alpha = ww / fmaxf(dn, 1e-9f);
    }
    const float4* fs4 = (const float4*)fs;
#pragma unroll
    for (int i = 0; i < 4; i++) {
        int f4i = i * 32 + lane;                   // 0..127 float4s
        int h = f4i >> 4;
        float a = __shfl(alpha, h, 32);
        float4 v = fs4[(size_t)s * 128 + f4i];
        float* ap = agg + (size_t)d * FDIM + f4i * 4;
        atomicAdd(ap + 0, v.x * a);
        atomicAdd(ap + 1, v.y * a);
        atomicAdd(ap + 2, v.z * a);
        atomicAdd(ap + 3, v.w * a);
    }
}

// h_out[n,d] = mean_h(agg[n,h,d] + bias[h,d]) ; optional relu
__global__ void node_finalize(const float* __restrict__ agg,
                              const float* __restrict__ bias,
                              float* __restrict__ hout, int relu) {
    int t = blockIdx.x * blockDim.x + threadIdx.x;
    if (t >= NNODES * DH) return;
    int n = t >> 6, d = t & 63;
    float sum = 0.f;
#pragma unroll
    for (int h = 0; h < NH; h++)
        sum += agg[(size_t)n * FDIM + h * DH + d] + bias[h * DH + d];
    float v = sum * (1.f / NH);
    if (relu) v = fmaxf(v, 0.f);
    hout[t] = v;
}

// per-graph mean readout accumulation
__global__ void readout(const float* __restrict__ h,
                        const int* __restrict__ gid,
                        float* __restrict__ gsum,
                        float* __restrict__ gcnt) {
    int t = blockIdx.x * blockDim.x + threadIdx.x;
    if (t >= NNODES * DH) return;
    int n = t >> 6, d = t & 63;
    int g = gid[n];
    atomicAdd(&gsum[(size_t)g * DH + d], h[t]);
    if (d == 0) atomicAdd(&gcnt[g], 1.f);
}

// head: relu(hg@Wh1+bh1)@Wh2+bh2 -> softmax ; one thread per graph
__global__ void head_softmax(const float* __restrict__ gsum,
                             const float* __restrict__ gcnt,
                             const float* __restrict__ Wh1,
                             const float* __restrict__ bh1,
                             const float* __restrict__ Wh2,
                             const float* __restrict__ bh2,
                             float* __restrict__ out) {
    int g = threadIdx.x;
    if (g >= NGRAPH) return;
    float inv = 1.f / fmaxf(gcnt[g], 1.f);
    float hid[DH];
#pragma unroll
    for (int j = 0; j < DH; j++) {
        float acc = bh1[j];
        for (int d = 0; d < DH; d++)
            acc += gsum[g * DH + d] * inv * Wh1[d * DH + j];
        hid[j] = fmaxf(acc, 0.f);
    }
    float lg[NC];
    float mx = -1e30f;
#pragma unroll
    for (int c = 0; c < NC; c++) {
        float acc = bh2[c];
#pragma unroll
        for (int j = 0; j < DH; j++) acc += hid[j] * Wh2[j * NC + c];
        lg[c] = acc;
        mx = fmaxf(mx, acc);
    }
    float sum = 0.f;
#pragma unroll
    for (int c = 0; c < NC; c++) { lg[c] = expf(lg[c] - mx); sum += lg[c]; }
    float is = 1.f / sum;
#pragma unroll
    for (int c = 0; c < NC; c++) out[g * NC + c] = lg[c] * is;
}

// -------------------------------------------------------------------------
extern "C" void kernel_launch(void* const* d_in, const int* in_sizes, int n_in,
                              void* d_out, int out_size, void* d_ws, size_t ws_size,
                              hipStream_t stream) {
    const float* X     = (const float*)d_in[0];
    const int*   src   = (const int*)  d_in[1];
    const int*   dst   = (const int*)  d_in[2];
    const int*   gid   = (const int*)  d_in[3];
    const float* W_in  = (const float*)d_in[4];
    const float* b_in  = (const float*)d_in[5];
    const float* Wsrc1 = (const float*)d_in[6];
    const float* bsrc1 = (const float*)d_in[7];
    const float* Wdst1 = (const float*)d_in[8];
    const float* bdst1 = (const float*)d_in[9];
    const float* attn1 = (const float*)d_in[10];
    const float* bias1 = (const float*)d_in[11];
    const float* Wsrc2 = (const float*)d_in[12];
    const float* bsrc2 = (const float*)d_in[13];
    const float* Wdst2 = (const float*)d_in[14];
    const float* bdst2 = (const float*)d_in[15];
    const float* attn2 = (const float*)d_in[16];
    const float* bias2 = (const float*)d_in[17];
    const float* Wh1   = (const float*)d_in[18];
    const float* bh1   = (const float*)d_in[19];
    const float* Wh2   = (const float*)d_in[20];
    const float* bh2   = (const float*)d_in[21];

    // ---- workspace layout (floats) ----
    float* ws = (float*)d_ws;
    size_t o = 0;
    float* h    = ws + o; o += (size_t)NNODES * DH;     // 1.28M  (h0, then h1, then h2-mean)
    float* fs   = ws + o; o += (size_t)NNODES * FDIM;   // 10.24M
    float* fd   = ws + o; o += (size_t)NNODES * FDIM;   // 10.24M
    float* wbuf = ws + o; o += (size_t)NEDGES * NH;     // 2.56M  (scores then exp-weights)
    float* m    = ws + o; o += (size_t)NNODES * NH;     // 0.16M  (uint-encoded then float)
    float* den  = ws + o; o += (size_t)NNODES * NH;     // 0.16M
    float* agg  = ws + o; o += (size_t)NNODES * FDIM;   // 10.24M
    float* gsum = ws + o; o += (size_t)NGRAPH * DH;
    float* gcnt = ws + o; o += (size_t)NGRAPH;

    const int TPB = 256;
    const int gemm_blk_h  = ((NNODES/16) * (DH/16)   + 7) / 8;   // 625
    const int gemm_blk_f  = ((NNODES/16) * (FDIM/16) + 7) / 8;   // 5000
    const int blkNH  = (NNODES * NH   + TPB - 1) / TPB;          // 625
    const int blkND  = (NNODES * DH   + TPB - 1) / TPB;          // 5000
    const int blkNF  = (NNODES * FDIM + TPB - 1) / TPB;          // 40000
    const int blkEH  = (NEDGES * NH   + TPB - 1) / TPB;          // 10000
    const int blkE   = (NEDGES + 7) / 8;                         // 40000

    // input projection: h0 = X @ W_in + b_in
    gemm_wmma_f32<<<gemm_blk_h, TPB, 0, stream>>>(X, W_in, b_in, h, NNODES, DIMIN, DH);

    for (int layer = 0; layer < 2; layer++) {
        const float* Ws = layer ? Wsrc2 : Wsrc1;
        const float* bs = layer ? bsrc2 : bsrc1;
        const float* Wd = layer ? Wdst2 : Wdst1;
        const float* bd = layer ? bdst2 : bdst1;
        const float* at = layer ? attn2 : attn1;
        const float* bi = layer ? bias2 : bias1;

        gemm_wmma_f32<<<gemm_blk_f, TPB, 0, stream>>>(h, Ws, bs, fs, NNODES, DH, FDIM);
        gemm_wmma_f32<<<gemm_blk_f, TPB, 0, stream>>>(h, Wd, bd, fd, NNODES, DH, FDIM);

        fill_f32<<<blkNH, TPB, 0, stream>>>(m,   0.f, NNODES * NH);  // 0 == encoded -inf sentinel
        fill_f32<<<blkNH, TPB, 0, stream>>>(den, 0.f, NNODES * NH);
        fill_f32<<<blkNF, TPB, 0, stream>>>(agg, 0.f, NNODES * FDIM);

        edge_scores<<<blkE, TPB, 0, stream>>>(fs, fd, at, src, dst, wbuf, (unsigned*)m);
        decode_m<<<blkNH, TPB, 0, stream>>>((unsigned*)m, NNODES * NH);
        edge_expw<<<blkEH, TPB, 0, stream>>>(wbuf, m, den, dst);
        edge_agg<<<blkE, TPB, 0, stream>>>(fs, wbuf, den, src, dst, agg);
        node_finalize<<<blkND, TPB, 0, stream>>>(agg, bi, h, /*relu=*/layer == 0);
    }

    // mean readout per graph
    fill_f32<<<(NGRAPH*DH + TPB-1)/TPB, TPB, 0, stream>>>(gsum, 0.f, NGRAPH * DH);
    fill_f32<<<1, TPB, 0, stream>>>(gcnt, 0.f, NGRAPH);
    readout<<<blkND, TPB, 0, stream>>>(h, gid, gsum, gcnt);

    // classification head + softmax
    head_softmax<<<1, 64, 0, stream>>>(gsum, gcnt, Wh1, bh1, Wh2, bh2, (float*)d_out);
}